// BboxDetectionHead_3745211483111
// MI455X (gfx1250) — compile-verified
//
#include <hip/hip_runtime.h>
#include <math.h>
#include <float.h>

// ---------------- problem constants (from reference) ----------------
#define THR      0.01f
#define IOU_THR  0.5f
#define MAX_DET  100
#define NCLS     90          // classes (also checked at runtime via in_sizes)

// ---------------- tiling ----------------
#define TILE        64       // anchors per block in phase 1 (64*90*4 = 22.5KB LDS tile)
#define K1_THREADS  256
#define CHUNK       256      // anchors per NMS chunk
#define MAX_CHUNKS  1024     // supports N <= 262144 (actual N = 196416 -> 768 chunks)
#define K2_THREADS  1024

// =====================================================================
// Phase 1: per-anchor class-max + box decode/clip.
// Classification rows are contiguous, so each block's tile is ONE
// contiguous span of TILE*C floats -> stage it into LDS with CDNA5
// async global->LDS b128 transfers, then reduce out of LDS.
// =====================================================================
__global__ void __launch_bounds__(K1_THREADS)
decode_score_kernel(const float* __restrict__ cls,
                    const float* __restrict__ reg,
                    const float* __restrict__ anc,
                    const int*   __restrict__ img_h_p,
                    const int*   __restrict__ img_w_p,
                    float* __restrict__ boxes,   // [N,4]
                    float* __restrict__ s,       // [N] thresholded scores
                    int N, int C)
{
    __shared__ float tile[TILE * NCLS];

    const int tid  = threadIdx.x;
    const int base = blockIdx.x * TILE;             // first anchor of this block

    const int totalF  = TILE * C;                   // floats wanted in tile
    const int nVec    = totalF >> 2;                // b128 transfers
    const float* gsrc = cls + (size_t)base * C;     // contiguous source span
    const int gLimitF = N * C - base * C;           // floats actually available

    // ---- async stage: global -> LDS (ASYNCcnt tracked) ----
    for (int v = tid; v < nVec; v += K1_THREADS) {
        const int f = v * 4;
        if (f + 3 < gLimitF) {
            unsigned ldsoff = (unsigned)(size_t)(void*)(&tile[f]); // low 32b of flat LDS addr
            const float* gp = gsrc + f;
            asm volatile("global_load_async_to_lds_b128 %0, %1, off"
                         :: "v"(ldsoff), "v"(gp) : "memory");
        }
    }
    // scalar tail (only if TILE*C not multiple of 4; 90*64 is, but stay general)
    for (int f = (nVec << 2) + tid; f < totalF; f += K1_THREADS)
        if (f < gLimitF) tile[f] = gsrc[f];

    asm volatile("s_wait_asynccnt 0x0" ::: "memory");  // my wave's async loads done
    __syncthreads();                                   // all waves' loads visible

    // ---- 4 lanes per anchor reduce the 90-class max out of LDS ----
    const int a    = tid >> 2;       // 0..TILE-1
    const int part = tid & 3;
    const int anchor = base + a;
    if (anchor < N) {
        const int c0 = (C * part) >> 2;
        const int c1 = (C * (part + 1)) >> 2;
        const float* row = &tile[a * C];
        float m = -FLT_MAX;
        for (int c = c0; c < c1; ++c) m = fmaxf(m, row[c]);
        m = fmaxf(m, __shfl_xor(m, 1));  // wave32 lane groups of 4
        m = fmaxf(m, __shfl_xor(m, 2));

        if (part == 0) {
            // BBoxTransform (std = .1,.1,.2,.2) + ClipBoxes
            const float ax1 = anc[anchor*4+0], ay1 = anc[anchor*4+1];
            const float ax2 = anc[anchor*4+2], ay2 = anc[anchor*4+3];
            const float wa = ax2 - ax1, ha = ay2 - ay1;
            const float cxa = ax1 + 0.5f*wa, cya = ay1 + 0.5f*ha;
            const float dx = reg[anchor*4+0]*0.1f, dy = reg[anchor*4+1]*0.1f;
            const float dw = reg[anchor*4+2]*0.2f, dh = reg[anchor*4+3]*0.2f;
            const float cx = cxa + dx*wa, cy = cya + dy*ha;
            const float w  = expf(dw)*wa, h  = expf(dh)*ha;
            const float W = (float)(*img_w_p), H = (float)(*img_h_p);
            float4 b;
            b.x = fminf(fmaxf(cx - 0.5f*w, 0.f), W);
            b.y = fminf(fmaxf(cy - 0.5f*h, 0.f), H);
            b.z = fminf(fmaxf(cx + 0.5f*w, 0.f), W);
            b.w = fminf(fmaxf(cy + 0.5f*h, 0.f), H);
            ((float4*)boxes)[anchor] = b;
            s[anchor] = (m > THR) ? m : -INFINITY;   // threshold mask like reference
        }
    }
}

// =====================================================================
// Phase 2: exact greedy NMS, single persistent workgroup (32 wave32s).
// Chunk maxima + chunk AABBs live in LDS; scores/boxes stream from L2.
// AABB test is a conservative exact filter (IoU>0.5 => intersection>0).
// =====================================================================
__global__ void __launch_bounds__(K2_THREADS)
nms_kernel(const float* __restrict__ cls,
           const float* __restrict__ boxes,  // [N,4]
           float* __restrict__ s,            // [N] live scores (mutated)
           float* __restrict__ out,          // 700 floats
           int N, int C)
{
    __shared__ float4 cbb [MAX_CHUNKS];   // chunk AABB (union of all boxes in chunk)
    __shared__ float  cmax[MAX_CHUNKS];   // chunk score max (live)
    __shared__ int    cidx[MAX_CHUNKS];   // chunk argmax (lowest index on ties)
    __shared__ float  rs[K2_THREADS];
    __shared__ int    ri[K2_THREADS];
    __shared__ int    pidx [MAX_DET];
    __shared__ float  pscore[MAX_DET];
    __shared__ int    pkeep[MAX_DET];
    __shared__ float4 pbox;

    const int tid    = threadIdx.x;
    const int nchunk = (N + CHUNK - 1) / CHUNK;   // assumed <= MAX_CHUNKS

    // ---- prologue: build chunk metadata (one full streaming pass) ----
    for (int c = tid; c < nchunk; c += K2_THREADS) {
        const int j0 = c * CHUNK, j1 = min(N, j0 + CHUNK);
        float bx1 = FLT_MAX, by1 = FLT_MAX, bx2 = -FLT_MAX, by2 = -FLT_MAX;
        float m = -INFINITY; int mi = j0;
        for (int j = j0; j < j1; ++j) {
            const float4 b = ((const float4*)boxes)[j];
            bx1 = fminf(bx1, b.x); by1 = fminf(by1, b.y);
            bx2 = fmaxf(bx2, b.z); by2 = fmaxf(by2, b.w);
            const float sv = s[j];
            if (sv > m) { m = sv; mi = j; }
        }
        cbb[c]  = make_float4(bx1, by1, bx2, by2);
        cmax[c] = m; cidx[c] = mi;
    }
    __syncthreads();

    for (int t = 0; t < MAX_DET; ++t) {
        // ---- block argmax over chunk maxima (ties -> lowest index) ----
        rs[tid] = (tid < nchunk) ? cmax[tid] : -INFINITY;
        ri[tid] = (tid < nchunk) ? cidx[tid] : 0x7fffffff;
        __syncthreads();
        for (int off = K2_THREADS / 2; off > 0; off >>= 1) {
            if (tid < off) {
                const float v2 = rs[tid + off]; const int i2 = ri[tid + off];
                if (v2 > rs[tid] || (v2 == rs[tid] && i2 < ri[tid])) { rs[tid] = v2; ri[tid] = i2; }
            }
            __syncthreads();
        }
        if (tid == 0) {
            const int pk = ri[0]; const float ps = rs[0];
            pidx[t] = pk; pscore[t] = ps; pkeep[t] = isfinite(ps) ? 1 : 0;
            pbox = ((const float4*)boxes)[pk];
        }
        __syncthreads();
        if (!pkeep[t]) continue;      // uniform: everything already -inf, suppression is a no-op

        const float4 P = pbox;
        const int pk = pidx[t];
        const int pchunk = pk / CHUNK;
        const float pa = (P.z - P.x) * (P.w - P.y);

        // ---- suppression: one chunk per thread; skip dead / non-overlapping ----
        for (int c = tid; c < nchunk; c += K2_THREADS) {
            if (cmax[c] == -INFINITY && c != pchunk) continue;
            const float4 B = cbb[c];
            const bool ov = (P.x < B.z) && (B.x < P.z) && (P.y < B.w) && (B.y < P.w);
            if (!ov && c != pchunk) continue;   // no positive intersection possible in chunk

            const int j0 = c * CHUNK, j1 = min(N, j0 + CHUNK);
            float m = -INFINITY; int mi = j0;
            for (int j = j0; j < j1; ++j) {
                float sv = s[j];
                if (sv != -INFINITY) {
                    const float4 b = ((const float4*)boxes)[j];
                    const float ix1 = fmaxf(P.x, b.x), iy1 = fmaxf(P.y, b.y);
                    const float ix2 = fminf(P.z, b.z), iy2 = fminf(P.w, b.w);
                    const float inter = fmaxf(ix2 - ix1, 0.f) * fmaxf(iy2 - iy1, 0.f);
                    const float a2 = (b.z - b.x) * (b.w - b.y);
                    const float iou = inter / (pa + a2 - inter + 1e-8f);
                    if (iou > IOU_THR || j == pk) { sv = -INFINITY; s[j] = -INFINITY; }
                }
                if (sv > m) { m = sv; mi = j; }
            }
            cmax[c] = m; cidx[c] = mi;
        }
        __syncthreads();
    }

    // ---- emit outputs: scores[100] | class[100] | boxes[100*4] | keep[100] ----
    for (int t = tid; t < MAX_DET; t += K2_THREADS) {
        const int keep = pkeep[t];
        const int idx  = pidx[t];
        out[t] = keep ? pscore[t] : 0.0f;
        if (keep) {
            const float* row = cls + (size_t)idx * C;
            float cm = -FLT_MAX; int ci = 0;
            for (int c = 0; c < C; ++c) { const float v = row[c]; if (v > cm) { cm = v; ci = c; } }
            out[MAX_DET + t] = (float)ci;
        } else {
            out[MAX_DET + t] = -1.0f;
        }
        float4 b = make_float4(0.f, 0.f, 0.f, 0.f);
        if (keep) b = ((const float4*)boxes)[idx];
        out[2*MAX_DET + t*4 + 0] = b.x;
        out[2*MAX_DET + t*4 + 1] = b.y;
        out[2*MAX_DET + t*4 + 2] = b.z;
        out[2*MAX_DET + t*4 + 3] = b.w;
        out[6*MAX_DET + t] = keep ? 1.0f : 0.0f;
    }
}

// =====================================================================
extern "C" void kernel_launch(void* const* d_in, const int* in_sizes, int n_in,
                              void* d_out, int out_size, void* d_ws, size_t ws_size,
                              hipStream_t stream) {
    const float* cls   = (const float*)d_in[0];   // [1,N,C]
    const float* reg   = (const float*)d_in[1];   // [1,N,4]
    const float* anc   = (const float*)d_in[2];   // [1,N,4]
    const int*   img_h = (const int*)d_in[3];     // scalar
    const int*   img_w = (const int*)d_in[4];     // scalar

    const int N = in_sizes[2] / 4;
    const int C = in_sizes[0] / N;                // 90

    float* boxes = (float*)d_ws;                  // N*4 floats
    float* s     = boxes + (size_t)N * 4;         // N floats

    const int blocks = (N + TILE - 1) / TILE;
    decode_score_kernel<<<blocks, K1_THREADS, 0, stream>>>(
        cls, reg, anc, img_h, img_w, boxes, s, N, C);
    nms_kernel<<<1, K2_THREADS, 0, stream>>>(
        cls, boxes, s, (float*)d_out, N, C);
}